// GCN_89172110999681
// MI455X (gfx1250) — compile-verified
//
#include <hip/hip_runtime.h>
#include <hip/hip_bf16.h>
#include <math.h>

// ---------------- problem constants (from reference) ----------------
#define N_NODES 50000
#define N_EDGES 800000
#define N_FEAT  512
#define N_HID   64
#define N_CLASS 40

typedef float v2f __attribute__((ext_vector_type(2)));
typedef float v4f __attribute__((ext_vector_type(4)));
typedef float v8f __attribute__((ext_vector_type(8)));

// ---------------- workspace layout (floats) ----------------
#define WS_XW     0
#define WS_H      3200000
#define WS_OUTPRE 6400000
#define WS_OUTACC 8400000

// ---------------- CDNA5 async global->LDS copy (ASYNCcnt path) ----------
#if defined(__HIP_DEVICE_COMPILE__) && defined(__gfx1250__) && \
    __has_builtin(__builtin_amdgcn_global_load_async_to_lds_b32) && \
    __has_builtin(__builtin_amdgcn_s_wait_asynccnt)
#define USE_ASYNC_LDS 1
#else
#define USE_ASYNC_LDS 0
#endif

__device__ __forceinline__ void async_g2l_b32(const float* g, float* l) {
#if USE_ASYNC_LDS
    __builtin_amdgcn_global_load_async_to_lds_b32(
        (__attribute__((address_space(1))) int*)(void*)g,
        (__attribute__((address_space(3))) int*)(void*)l, 0, 0);
#else
    *l = *g;
#endif
}

__device__ __forceinline__ void wait_async_and_sync() {
#if USE_ASYNC_LDS
    __builtin_amdgcn_s_wait_asynccnt(0);   // my wave's async copies done
#endif
    __syncthreads();                       // everyone's copies visible
}

__device__ __forceinline__ void atomic_add_f32(float* p, float v) {
#if defined(__gfx1250__)
    unsafeAtomicAdd(p, v);   // native global_atomic_add_f32
#else
    atomicAdd(p, v);
#endif
}

// ================= GEMM1: XW = x @ W1  (f32 WMMA 16x16x4) =================
// block = 256 threads = 8 waves; each wave computes a 16-row x 64-col strip.
// W1 staged TRANSPOSED in LDS via async-to-LDS (per-lane LDS address ->
// transpose comes free), K-chunks of 128 so B frag is one ds_load_b64.
#define KCHUNK 128
#define LDS_S1 130   // 128 + 2 pad floats (8B-aligned rows, spreads banks)

__global__ __launch_bounds__(256) void gcn_gemm1(const float* __restrict__ x,
                                                 const float* __restrict__ w1,
                                                 float* __restrict__ xw) {
    __shared__ float w1t[N_HID * LDS_S1];   // 64*130*4 = 33,280 B
    const int tid  = threadIdx.x;
    const int lane = tid & 31;
    const int wave = tid >> 5;
    const int half = lane >> 4;   // 0: K,K+1   1: K+2,K+3
    const int l16  = lane & 15;
    const int row_base = blockIdx.x * 128 + wave * 16;

    int m = row_base + l16;
    if (m >= N_NODES) m = N_NODES - 1;   // clamp: keeps EXEC all-ones for WMMA
    const float* xrow = x + (size_t)m * N_FEAT;

    v8f acc0 = {}, acc1 = {}, acc2 = {}, acc3 = {};

    for (int kb = 0; kb < N_FEAT; kb += KCHUNK) {
        __syncthreads();   // protect previous chunk's readers
        // stage W1[kb..kb+127][0..63] transposed: w1t[c][k_local]
        for (int i = tid; i < N_HID * KCHUNK; i += 256) {
            int kl = i >> 6;       // 0..127
            int c  = i & 63;       // coalesced global access along c
            async_g2l_b32(w1 + (size_t)(kb + kl) * N_HID + c,
                          &w1t[c * LDS_S1 + kl]);
        }
        wait_async_and_sync();

        #pragma unroll 4
        for (int kl = 0; kl < KCHUNK; kl += 4) {
            // A fragment (16x4 f32): one b64 load per lane
            v2f a = *(const v2f*)(xrow + kb + kl + half * 2);
            // B fragments (4x16 f32) from transposed LDS: one ds_load_b64 each
            const int kk = kl + half * 2;
            v2f b0 = *(const v2f*)(&w1t[( 0 + l16) * LDS_S1 + kk]);
            v2f b1 = *(const v2f*)(&w1t[(16 + l16) * LDS_S1 + kk]);
            v2f b2 = *(const v2f*)(&w1t[(32 + l16) * LDS_S1 + kk]);
            v2f b3 = *(const v2f*)(&w1t[(48 + l16) * LDS_S1 + kk]);
            acc0 = __builtin_amdgcn_wmma_f32_16x16x4_f32(false, a, false, b0, (short)0, acc0, false, false);
            acc1 = __builtin_amdgcn_wmma_f32_16x16x4_f32(false, a, false, b1, (short)0, acc1, false, false);
            acc2 = __builtin_amdgcn_wmma_f32_16x16x4_f32(false, a, false, b2, (short)0, acc2, false, false);
            acc3 = __builtin_amdgcn_wmma_f32_16x16x4_f32(false, a, false, b3, (short)0, acc3, false, false);
        }
    }

    // D layout: VGPR v, lanes0-15 -> M=v, lanes16-31 -> M=v+8; N = l16 + 16*ct
    const int mrow = row_base + half * 8;
    v8f accs[4] = {acc0, acc1, acc2, acc3};
    #pragma unroll
    for (int ct = 0; ct < 4; ++ct) {
        const int n = ct * 16 + l16;
        #pragma unroll
        for (int v = 0; v < 8; ++v) {
            int mm = mrow + v;
            if (mm < N_NODES) xw[(size_t)mm * N_HID + n] = accs[ct][v];
        }
    }
}

// ================= zero accumulators =================
__global__ __launch_bounds__(256) void gcn_zero(float* __restrict__ h,
                                                float* __restrict__ oacc) {
    size_t i = (size_t)blockIdx.x * 256 + threadIdx.x;
    if (i < (size_t)N_NODES * N_HID)   h[i]    = 0.0f;
    if (i < (size_t)N_NODES * N_CLASS) oacc[i] = 0.0f;
}

// ===== SpMM1: h[dst] += vals * xw[src]; 4 feats/thread (b128 gather) ======
__global__ __launch_bounds__(256) void gcn_scatter64(const float* __restrict__ xw,
                                                     const float* __restrict__ vals,
                                                     const int*   __restrict__ src,
                                                     const int*   __restrict__ dst,
                                                     float* __restrict__ h) {
    size_t i = (size_t)blockIdx.x * 256 + threadIdx.x;
    if (i >= (size_t)N_EDGES * (N_HID / 4)) return;
    int e = (int)(i >> 4);            // 16 groups of 4 feats per edge
    int f = (int)(i & 15) << 2;
    int s = src[e], d = dst[e];
    float w = vals[e];
    v4f t = *(const v4f*)(xw + (size_t)s * N_HID + f);   // global_load_b128
    float* hp = h + (size_t)d * N_HID + f;
    atomic_add_f32(hp + 0, w * t.x);
    atomic_add_f32(hp + 1, w * t.y);
    atomic_add_f32(hp + 2, w * t.z);
    atomic_add_f32(hp + 3, w * t.w);
}

// ================= GEMM2: out_pre = relu(h + b1) @ W2  (N padded 40->48) ==
#define LDS_S2 66    // 64 + 2 pad

__global__ __launch_bounds__(256) void gcn_gemm2(const float* __restrict__ h,
                                                 const float* __restrict__ b1,
                                                 const float* __restrict__ w2,
                                                 float* __restrict__ out_pre) {
    __shared__ float w2t[48 * LDS_S2];  // 12,672 B, transposed + zero-padded
    __shared__ float b1s[N_HID];
    const int tid = threadIdx.x;
    for (int i = tid; i < 48 * N_HID; i += 256) {
        int c = i >> 6;     // 0..47
        int k = i & 63;
        if (c < N_CLASS) {
            async_g2l_b32(w2 + (size_t)k * N_CLASS + c, &w2t[c * LDS_S2 + k]);
        } else {
            w2t[c * LDS_S2 + k] = 0.0f;
        }
    }
    if (tid < N_HID) b1s[tid] = b1[tid];
    wait_async_and_sync();

    const int lane = tid & 31;
    const int wave = tid >> 5;
    const int half = lane >> 4;
    const int l16  = lane & 15;
    const int row_base = blockIdx.x * 128 + wave * 16;

    int m = row_base + l16;
    if (m >= N_NODES) m = N_NODES - 1;
    const float* hrow = h + (size_t)m * N_HID;

    v8f acc0 = {}, acc1 = {}, acc2 = {};
    #pragma unroll 4
    for (int k = 0; k < N_HID; k += 4) {
        const int kk = k + half * 2;
        v2f a = *(const v2f*)(hrow + kk);
        a.x = fmaxf(a.x + b1s[kk], 0.0f);       // fused bias + ReLU
        a.y = fmaxf(a.y + b1s[kk + 1], 0.0f);
        v2f b0  = *(const v2f*)(&w2t[( 0 + l16) * LDS_S2 + kk]);
        v2f b1f = *(const v2f*)(&w2t[(16 + l16) * LDS_S2 + kk]);
        v2f b2f = *(const v2f*)(&w2t[(32 + l16) * LDS_S2 + kk]);
        acc0 = __builtin_amdgcn_wmma_f32_16x16x4_f32(false, a, false, b0,  (short)0, acc0, false, false);
        acc1 = __builtin_amdgcn_wmma_f32_16x16x4_f32(false, a, false, b1f, (short)0, acc1, false, false);
        acc2 = __builtin_amdgcn_wmma_f32_16x16x4_f32(false, a, false, b2f, (short)0, acc2, false, false);
    }

    const int mrow = row_base + half * 8;
    v8f accs[3] = {acc0, acc1, acc2};
    #pragma unroll
    for (int ct = 0; ct < 3; ++ct) {
        const int n = ct * 16 + l16;
        if (n >= N_CLASS) continue;
        #pragma unroll
        for (int v = 0; v < 8; ++v) {
            int mm = mrow + v;
            if (mm < N_NODES) out_pre[(size_t)mm * N_CLASS + n] = accs[ct][v];
        }
    }
}

// ===== SpMM2: out_acc[dst] += vals * out_pre[src]; 4 feats/thread =========
__global__ __launch_bounds__(256) void gcn_scatter40(const float* __restrict__ op,
                                                     const float* __restrict__ vals,
                                                     const int*   __restrict__ src,
                                                     const int*   __restrict__ dst,
                                                     float* __restrict__ oacc) {
    size_t i = (size_t)blockIdx.x * 256 + threadIdx.x;
    if (i >= (size_t)N_EDGES * (N_CLASS / 4)) return;
    int e = (int)(i / 10);            // 10 groups of 4 feats per edge
    int f = (int)(i - (size_t)e * 10) << 2;
    int s = src[e], d = dst[e];
    float w = vals[e];
    v4f t = *(const v4f*)(op + (size_t)s * N_CLASS + f);  // 16B-aligned (160B rows)
    float* pp = oacc + (size_t)d * N_CLASS + f;
    atomic_add_f32(pp + 0, w * t.x);
    atomic_add_f32(pp + 1, w * t.y);
    atomic_add_f32(pp + 2, w * t.z);
    atomic_add_f32(pp + 3, w * t.w);
}

// ================= bias + log_softmax: one wave32 per row =================
__global__ __launch_bounds__(256) void gcn_logsoftmax(const float* __restrict__ oacc,
                                                      const float* __restrict__ b2,
                                                      float* __restrict__ out) {
    const int tid  = threadIdx.x;
    const int lane = tid & 31;
    const int wave = tid >> 5;
    const int row  = blockIdx.x * 8 + wave;
    if (row >= N_NODES) return;
    const float* r = oacc + (size_t)row * N_CLASS;

    float z1 = r[lane] + b2[lane];                                      // f = lane (<40)
    float z2 = (lane < 8) ? (r[lane + 32] + b2[lane + 32]) : -INFINITY; // f = lane+32

    float lmax = fmaxf(z1, z2);
    #pragma unroll
    for (int off = 16; off > 0; off >>= 1)
        lmax = fmaxf(lmax, __shfl_xor(lmax, off, 32));

    float ls = expf(z1 - lmax) + ((lane < 8) ? expf(z2 - lmax) : 0.0f);
    #pragma unroll
    for (int off = 16; off > 0; off >>= 1)
        ls += __shfl_xor(ls, off, 32);

    const float logZ = lmax + logf(ls);
    out[(size_t)row * N_CLASS + lane] = z1 - logZ;
    if (lane < 8) out[(size_t)row * N_CLASS + 32 + lane] = z2 - logZ;
}

// =========================== launch ===========================
extern "C" void kernel_launch(void* const* d_in, const int* in_sizes, int n_in,
                              void* d_out, int out_size, void* d_ws, size_t ws_size,
                              hipStream_t stream) {
    const float* x    = (const float*)d_in[0];
    const float* av   = (const float*)d_in[1];
    const float* W1   = (const float*)d_in[2];
    const float* b1   = (const float*)d_in[3];
    const float* W2   = (const float*)d_in[4];
    const float* b2   = (const float*)d_in[5];
    const int*   src  = (const int*)d_in[6];
    const int*   dst  = (const int*)d_in[7];
    float*       out  = (float*)d_out;

    float* ws      = (float*)d_ws;
    float* xw      = ws + WS_XW;
    float* h       = ws + WS_H;
    float* out_pre = ws + WS_OUTPRE;
    float* out_acc = ws + WS_OUTACC;

    const int gemm_blocks = (N_NODES + 127) / 128;                              // 391
    const int zero_blocks = ((N_NODES * N_HID) + 255) / 256;                    // 12,500
    const int sc1_blocks  = (int)(((size_t)N_EDGES * (N_HID / 4)  + 255) / 256); // 50,000
    const int sc2_blocks  = (int)(((size_t)N_EDGES * (N_CLASS / 4) + 255) / 256);// 31,250
    const int sm_blocks   = (N_NODES + 7) / 8;                                  // 6,250

    gcn_zero      <<<zero_blocks, 256, 0, stream>>>(h, out_acc);
    gcn_gemm1     <<<gemm_blocks, 256, 0, stream>>>(x, W1, xw);
    gcn_scatter64 <<<sc1_blocks,  256, 0, stream>>>(xw, av, src, dst, h);
    gcn_gemm2     <<<gemm_blocks, 256, 0, stream>>>(h, b1, W2, out_pre);
    gcn_scatter40 <<<sc2_blocks,  256, 0, stream>>>(out_pre, av, src, dst, out_acc);
    gcn_logsoftmax<<<sm_blocks,   256, 0, stream>>>(out_acc, b2, out);
}